// QEncoder_13898514170709
// MI455X (gfx1250) — compile-verified
//
#include <hip/hip_runtime.h>
#include <math.h>

#define DIM    128
#define BATCH  2048
#define NT     128          // Chebyshev terms (covers |z| = |alpha|*L up to ~100)
#define LCHEB  23.0f        // spectral bound of i(adag - a), 2*sqrt(127) ~ 22.54
#define SQ_SCALE (1.0f/32768.0f)   // 2^-15 scaling for expm scaling-and-squaring

typedef __attribute__((ext_vector_type(2))) float v2f;
typedef __attribute__((ext_vector_type(8))) float v8f;

// ===========================================================================
// Kernel 1: ONE workgroup (1024 threads = 32 waves) computes
//   E = expm(50*(a^2 - adag^2))  via scaling-and-squaring, entirely in LDS:
//     - Taylor order 10 by Horner; generator is pentadiagonal -> banded O(N^2)
//     - 15 squarings as 128x128x128 fp32 WMMA matmuls (2 tiles per wave)
//   then the Chebyshev basis s_k of K = adag - a applied to v = E[:,0]:
//     s_{k+1} = (2/L) K s_k + s_{k-1};   smat[m*NT+k] = s_k[m]
// ===========================================================================
__global__ void __launch_bounds__(1024)
expm_cheb_kernel(float* __restrict__ smat) {
    __shared__ float Mb[2][DIM * DIM];   // 2 x 64KB ping-pong
    __shared__ float sp[DIM];
    __shared__ float sc[DIM];

    const int tid  = threadIdx.x;
    const int lane = tid & 31;
    const int wave = tid >> 5;

    // ---- seed: T = I + Bs/10 ----
    for (int i = tid; i < DIM * DIM; i += 1024) {
        int n = i >> 7, c = i & 127;
        float bs = 0.0f;
        if (c == n + 2) bs =  50.0f * sqrtf((float)((n + 1) * (n + 2))) * SQ_SCALE;
        if (c + 2 == n) bs = -50.0f * sqrtf((float)(n * (n - 1)))       * SQ_SCALE;
        Mb[0][i] = bs * 0.1f + ((n == c) ? 1.0f : 0.0f);
    }
    __syncthreads();

    int cs = 0;
    // ---- Horner: T <- I + (Bs*T)/j, j=9..1 ; Bs is pentadiagonal ----
    for (int j = 9; j >= 1; --j) {
        float invj = 1.0f / (float)j;
        for (int i = tid; i < DIM * DIM; i += 1024) {
            int n = i >> 7, c = i & 127;
            float acc = 0.0f;
            if (n <= DIM - 3)
                acc += 50.0f * sqrtf((float)((n + 1) * (n + 2))) * SQ_SCALE * Mb[cs][(n + 2) * DIM + c];
            if (n >= 2)
                acc -= 50.0f * sqrtf((float)(n * (n - 1))) * SQ_SCALE * Mb[cs][(n - 2) * DIM + c];
            Mb[cs ^ 1][i] = acc * invj + ((n == c) ? 1.0f : 0.0f);
        }
        __syncthreads();
        cs ^= 1;
    }

    // ---- 15 squarings: E <- E*E via V_WMMA_F32_16X16X4_F32, LDS operands ----
    for (int sq = 0; sq < 15; ++sq) {
        for (int tt = 0; tt < 2; ++tt) {
            int tile = wave + tt * 32;          // 64 tiles of 16x16
            int tm = tile >> 3, tn = tile & 7;
            int r  = lane & 15, hi = lane >> 4;
            int m  = tm * 16 + r;
            int n  = tn * 16 + r;
            v8f acc = {};
            for (int k0 = 0; k0 < DIM; k0 += 4) {
                v2f a, b;
                a.x = Mb[cs][m * DIM + k0 + 2 * hi];
                a.y = Mb[cs][m * DIM + k0 + 2 * hi + 1];
                b.x = Mb[cs][(k0 + 2 * hi) * DIM + n];
                b.y = Mb[cs][(k0 + 2 * hi + 1) * DIM + n];
                acc = __builtin_amdgcn_wmma_f32_16x16x4_f32(false, a, false, b,
                                                            (short)0, acc, false, false);
            }
            for (int rr = 0; rr < 8; ++rr) {
                int row = tm * 16 + rr + hi * 8;
                Mb[cs ^ 1][row * DIM + n] = acc[rr];
            }
        }
        __syncthreads();
        cs ^= 1;
    }

    // ---- Chebyshev basis from v = E[:,0] (all threads hit every barrier) ----
    int n = tid;
    float cUp = 0.f, cDn = 0.f;
    if (n < DIM) {
        cUp = sqrtf((float)(n + 1));
        cDn = (n > 0) ? sqrtf((float)n) : 0.f;
        sp[n] = Mb[cs][n * DIM + 0];
    }
    __syncthreads();
    if (n < DIM) {
        smat[n * NT + 0] = sp[n];
        float ks = ((n > 0) ? cDn * sp[n - 1] : 0.f)
                 - ((n < DIM - 1) ? cUp * sp[n + 1] : 0.f);
        sc[n] = ks * (1.0f / LCHEB);
    }
    __syncthreads();
    if (n < DIM) smat[n * NT + 1] = sc[n];

    for (int k = 2; k < NT; ++k) {
        float nx = 0.f;
        if (n < DIM) {
            float t = ((n > 0) ? cDn * sc[n - 1] : 0.f)
                    - ((n < DIM - 1) ? cUp * sc[n + 1] : 0.f);
            nx = (2.0f / LCHEB) * t + sp[n];
        }
        __syncthreads();
        if (n < DIM) { sp[n] = sc[n]; sc[n] = nx; }
        __syncthreads();
        if (n < DIM) smat[n * NT + k] = nx;
    }
}

// ===========================================================================
// Kernel 2: fused per-batch pipeline. One block = 16 output columns.
//   stage 1: 16 threads -> Miller Bessel recurrence, W[k][c] in LDS
//   stage 2: 8 waves    -> Out16x16 tiles = smat x W via WMMA (B from LDS)
//   stage 3: per-column L2 normalize + pack complex64
// ===========================================================================
__global__ void __launch_bounds__(256)
batch_kernel(const float* __restrict__ inputs, const float* __restrict__ smat,
             float* __restrict__ out) {
    __shared__ float Wl[NT * 16];    // coefficients [k][c]
    __shared__ float Ol[DIM * 16];   // output tile  [m][c]
    __shared__ float rsn[16];

    const int tid  = threadIdx.x;
    const int lane = tid & 31;
    const int wave = tid >> 5;
    const int b0   = blockIdx.x * 16;

    // ---- stage 1: Chebyshev coefficients w_k = (2-d_k0) J_k(alpha*L) ----
    if (tid < 16) {
        int   b  = b0 + tid;
        float z  = 0.5f * inputs[b] * LCHEB;
        float az = fabsf(z);
        if (az < 1e-4f) {
            for (int k = 0; k < NT; ++k) Wl[k * 16 + tid] = 0.f;
            Wl[0 * 16 + tid] = 1.0f;
            Wl[1 * 16 + tid] = z;              // 2 * J1(z) ~ z
        } else {
            const int M0 = 168;
            float jp = 0.f, jc = 1e-30f, sum = 2e-30f;
            float tia = 2.0f / az;
            for (int k = M0; k >= 1; --k) {
                float jm = ((float)k * tia) * jc - jp;
                jp = jc; jc = jm;
                int kk = k - 1;
                if (kk < NT) Wl[kk * 16 + tid] = jc;
                if (kk == 0) sum += jc;
                else if ((kk & 1) == 0) sum += 2.0f * jc;
                if (fabsf(jc) > 1e18f) {
                    const float s = 1e-18f;
                    jc *= s; jp *= s; sum *= s;
                    int st = kk < 0 ? 0 : kk;
                    if (st < NT)
                        for (int m2 = st; m2 < NT; ++m2) Wl[m2 * 16 + tid] *= s;
                }
            }
            if (fabsf(sum) < 1e-30f) sum = 1e-30f;
            float inv = 1.0f / sum;            // J0 + 2*sum(J_even) = 1
            for (int k = 0; k < NT; ++k) {
                float sgn = ((z < 0.f) && (k & 1)) ? -1.f : 1.f;  // J_k(-z)=(-1)^k J_k(z)
                float f   = (k == 0) ? 1.0f : 2.0f;
                Wl[k * 16 + tid] *= inv * sgn * f;
            }
        }
    }
    __syncthreads();

    // ---- stage 2: wave `tm` computes rows [tm*16, tm*16+16) x 16 cols ----
    {
        int tm = wave;
        int r  = lane & 15, hi = lane >> 4;
        int m  = tm * 16 + r;
        v8f acc = {};
        for (int k0 = 0; k0 < NT; k0 += 4) {
            v2f a, bv;
            a.x  = smat[m * NT + k0 + 2 * hi];
            a.y  = smat[m * NT + k0 + 2 * hi + 1];
            bv.x = Wl[(k0 + 2 * hi) * 16 + r];
            bv.y = Wl[(k0 + 2 * hi + 1) * 16 + r];
            acc = __builtin_amdgcn_wmma_f32_16x16x4_f32(false, a, false, bv,
                                                        (short)0, acc, false, false);
        }
        for (int rr = 0; rr < 8; ++rr) {
            int row = tm * 16 + rr + hi * 8;
            Ol[row * 16 + r] = acc[rr];
        }
    }
    __syncthreads();

    // ---- stage 3: per-column L2 norm ----
    if (tid < 16) {
        float s = 0.f;
        for (int m = 0; m < DIM; ++m) { float v = Ol[m * 16 + tid]; s += v * v; }
        rsn[tid] = rsqrtf(fmaxf(s, 1e-30f));
    }
    __syncthreads();

    // ---- pack complex64 (imag = 0): 8 values per thread ----
    int c  = tid & 15;
    int m0 = tid >> 4;
    int b  = b0 + c;
    float rs = rsn[c];
    for (int i = 0; i < 8; ++i) {
        int m = m0 + 16 * i;
        float v = Ol[m * 16 + c] * rs;
        out[(b * DIM + m) * 2 + 0] = v;
        out[(b * DIM + m) * 2 + 1] = 0.0f;
    }
}

// ===========================================================================
extern "C" void kernel_launch(void* const* d_in, const int* in_sizes, int n_in,
                              void* d_out, int out_size, void* d_ws, size_t ws_size,
                              hipStream_t stream) {
    (void)in_sizes; (void)n_in; (void)out_size; (void)ws_size;
    const float* inputs = (const float*)d_in[0];   // [2048] float32
    // d_in[1] = vacuum_state (implicit |0> -- handled analytically)
    float* out  = (float*)d_out;                   // [2048,128] complex64 interleaved
    float* smat = (float*)d_ws;                    // 128*NT floats (64KB)

    expm_cheb_kernel<<<1, 1024, 0, stream>>>(smat);
    batch_kernel<<<BATCH / 16, 256, 0, stream>>>(inputs, smat, out);
}